// EDN_Model_18811956757062
// MI455X (gfx1250) — compile-verified
//
#include <hip/hip_runtime.h>
#include <hip/hip_bf16.h>

// ---------------------------------------------------------------------------
// EDN equivariant GNN, fused for MI455X (gfx1250, wave32, WMMA).
// Radial MLPs run as v_wmma_f32_16x16x4_f32 GEMMs per 16-edge tile; the
// per-edge R matrices never touch HBM (LDS bounce only). Scatter via
// global_atomic_add_f32. W1 is staged zero-padded in LDS so WMMA operand
// feeds are branchless (no exec-mask divergence in the GEMM loop).
// ---------------------------------------------------------------------------

#define NNODES 16384
#define NEDGES 196608
#define NCA    2048
#define MULC   12
#define HID    12

typedef float v2f __attribute__((ext_vector_type(2)));
typedef float v8f __attribute__((ext_vector_type(8)));

__device__ __forceinline__ float sspf(float v) {           // shifted softplus
  float sp = (v > 20.0f) ? v : log1pf(__expf(v));
  return sp - 0.69314718055994531f;
}
__device__ __forceinline__ float eluf(float v) {
  return v > 0.0f ? v : (__expf(v) - 1.0f);
}
__device__ __forceinline__ float epsf(int i, int j, int k) {
  if (i == j || j == k || i == k) return 0.0f;
  return ((i==0&&j==1)||(i==1&&j==2)||(i==2&&j==0)) ? 1.0f : -1.0f;
}

#define INV_DEG 0.28867513459481287f   // 1/sqrt(E/N)

// ---- workspace layout (float offsets) -------------------------------------
constexpr size_t OFF_CG   = 0;                                  // 640
constexpr size_t OFF_FEAT = 640;                                // N*12
constexpr size_t OFF_A0   = OFF_FEAT + (size_t)NNODES*12;
constexpr size_t OFF_A1   = OFF_A0   + (size_t)NNODES*12;
constexpr size_t OFF_A2   = OFF_A1   + (size_t)NNODES*36;
constexpr size_t OFF_M0   = OFF_A2   + (size_t)NNODES*60;
constexpr size_t OFF_M1   = OFF_M0   + (size_t)NNODES*12;
constexpr size_t OFF_M2   = OFF_M1   + (size_t)NNODES*36;
constexpr size_t OFF_B0   = OFF_M2   + (size_t)NNODES*60;
constexpr size_t OFF_B1   = OFF_B0   + (size_t)NNODES*36;
constexpr size_t OFF_B2   = OFF_B1   + (size_t)NNODES*216;

// ---- path metadata for CG setup kernel ------------------------------------
__constant__ int c_PI[15]    = {0,0,0,1,1,1,1,1,1,2,2,2,2,2,2};
__constant__ int c_PF[15]    = {0,1,2,0,1,1,1,2,2,0,1,1,2,2,2};
__constant__ int c_PO[15]    = {0,1,2,1,0,1,2,1,2,2,1,2,0,1,2};
__constant__ int c_CGOFF[15] = {0,1,10,35,44,53,80,125,170,245,270,315,390,415,490};

// ===========================================================================
// CG setup: closed-form SO(3) invariant tensors, unit Frobenius norm,
// sign fixed by first argmax |element| (matches reference convention).
// ===========================================================================
__global__ void k_cg_setup(float* __restrict__ CGout)
{
  if (threadIdx.x != 0 || blockIdx.x != 0) return;
  const float c2a = 0.5462742152960396f;     // = c2b/2
  const float c2c = 0.31539156525252005f;
  float M[5][3][3];
  for (int a = 0; a < 5; ++a)
    for (int i = 0; i < 3; ++i)
      for (int j = 0; j < 3; ++j) M[a][i][j] = 0.0f;
  M[0][0][0]= c2a; M[0][1][1]=-c2a;
  M[1][0][1]= c2a; M[1][1][0]= c2a;
  M[2][1][2]= c2a; M[2][2][1]= c2a;
  M[3][0][0]=-c2c; M[3][1][1]=-c2c; M[3][2][2]=2.0f*c2c;
  M[4][0][2]= c2a; M[4][2][0]= c2a;

  for (int p = 0; p < 15; ++p) {
    int li = c_PI[p], lf = c_PF[p], lo = c_PO[p];
    int ni = 2*li+1, nf = 2*lf+1, no = 2*lo+1;
    int code = li*100 + lf*10 + lo;
    float T[125];
    for (int a = 0; a < ni; ++a)
      for (int b = 0; b < nf; ++b)
        for (int c = 0; c < no; ++c) {
          float v = 0.0f;
          switch (code) {
            case 0:   v = 1.0f; break;
            case 11: case 22:   v = (b==c) ? 1.0f : 0.0f; break;  // (0,l,l)
            case 101: case 202: v = (a==c) ? 1.0f : 0.0f; break;  // (l,0,l)
            case 110: case 220: v = (a==b) ? 1.0f : 0.0f; break;  // (l,l,0)
            case 111: v = epsf(a,b,c); break;
            case 112: v = M[c][a][b]; break;
            case 121: v = M[b][a][c]; break;
            case 211: v = M[a][b][c]; break;
            case 122: { float acc=0;
              for (int j=0;j<3;++j) for (int k=0;k<3;++k) { float e=epsf(a,j,k);
                if (e!=0.0f){ float mm=0; for (int l2=0;l2<3;++l2) mm+=M[b][j][l2]*M[c][l2][k]; acc+=e*mm; } }
              v=acc; break; }
            case 212: { float acc=0;
              for (int j=0;j<3;++j) for (int k=0;k<3;++k) { float e=epsf(b,j,k);
                if (e!=0.0f){ float mm=0; for (int l2=0;l2<3;++l2) mm+=M[a][j][l2]*M[c][l2][k]; acc+=e*mm; } }
              v=acc; break; }
            case 221: { float acc=0;
              for (int j=0;j<3;++j) for (int k=0;k<3;++k) { float e=epsf(c,j,k);
                if (e!=0.0f){ float mm=0; for (int l2=0;l2<3;++l2) mm+=M[a][j][l2]*M[b][l2][k]; acc+=e*mm; } }
              v=acc; break; }
            case 222: { float acc=0;
              for (int j=0;j<3;++j) for (int k=0;k<3;++k) for (int l2=0;l2<3;++l2)
                acc += M[a][j][k]*M[b][k][l2]*M[c][l2][j];
              v=acc; break; }
          }
          T[(a*nf + b)*no + c] = v;
        }
    int sz = ni*nf*no;
    float nrm = 0.0f;
    for (int t = 0; t < sz; ++t) nrm += T[t]*T[t];
    nrm = sqrtf(nrm);
    int amax = 0; float mx = -1.0f;
    for (int t = 0; t < sz; ++t) { float a = fabsf(T[t]); if (a > mx) { mx = a; amax = t; } }
    float inv = ((T[amax] < 0.0f) ? -1.0f : 1.0f) / nrm;
    for (int t = 0; t < sz; ++t) CGout[c_CGOFF[p] + t] = T[t]*inv;
  }
}

// ===========================================================================
// lin1: feat[n][u] = x[n][0:4] @ lin1_W
// ===========================================================================
__global__ void k_lin1(const float* __restrict__ x, const float* __restrict__ W,
                       float* __restrict__ feat)
{
  int idx = blockIdx.x*blockDim.x + threadIdx.x;
  if (idx >= NNODES*MULC) return;
  int n = idx / MULC, u = idx % MULC;
  float acc = 0.0f;
  #pragma unroll
  for (int c = 0; c < 4; ++c) acc += x[n*4 + c]*W[c*MULC + u];
  feat[idx] = acc;
}

// ===========================================================================
// Radial MLP for a 16-edge tile via fp32 WMMA 16x16x4:
//   GEMM1: basis[16xNB] @ W1[NBx12] (+bias, relu)  -> h  (LDS)
//   GEMM2: h[16x12] @ W2[12x144]                   -> R  (LDS, [edge][u*12+v])
// One wave (32 lanes), EXEC all ones. K dims are exact multiples of 4.
// W1 staged zero-padded (NBx16) in LDS -> branchless B-fragment feed.
// ===========================================================================
__device__ __forceinline__ void radial_to_lds(
    int lane, float r, int NB, float rmax,
    const float* __restrict__ W1, const float* __restrict__ bias,
    const float* __restrict__ W2,
    float* ldsW1, float* ldsH, float* ldsR)
{
  int lm = lane & 15, lhi = (lane >> 4) & 1;

  // stage W1 (NBx12) into LDS as NBx16 with zero-padded columns (branchless)
  for (int idx = lane; idx < NB*16; idx += 32) {
    int row = idx >> 4, col = idx & 15;
    float w = W1[row*12 + (col < 12 ? col : 11)];   // unconditional load
    ldsW1[idx] = (col < 12) ? w : 0.0f;             // v_cndmask, no branch
  }
  __syncthreads();

  float sigma = rmax / (float)NB;
  float cstep = rmax / (float)(NB - 1);
  float isig  = 1.0f / sigma;

  v8f h = {};
  for (int kk = 0; kk < NB; kk += 4) {
    v2f a, b;
    #pragma unroll
    for (int j = 0; j < 2; ++j) {
      int k = kk + 2*lhi + j;                       // A(m=lm, k), B(k, n=lm)
      float d = (r - cstep*(float)k) * isig;
      a[j] = __expf(-0.5f*d*d);
      b[j] = ldsW1[k*16 + lm];
    }
    h = __builtin_amdgcn_wmma_f32_16x16x4_f32(false, a, false, b, (short)0, h, false, false);
  }
  float bn = bias[lm < HID ? lm : HID-1];
  if (lm >= HID) bn = 0.0f;
  #pragma unroll
  for (int g = 0; g < 8; ++g) {                      // C(m=g+8*lhi, n=lm)
    float hv = fmaxf(h[g] + bn, 0.0f);
    if (lm >= HID) hv = 0.0f;
    ldsH[(g + 8*lhi)*16 + lm] = hv;
  }
  __syncthreads();

  v2f aK[3];
  #pragma unroll
  for (int kk3 = 0; kk3 < 3; ++kk3)
    #pragma unroll
    for (int j = 0; j < 2; ++j)
      aK[kk3][j] = ldsH[lm*16 + kk3*4 + 2*lhi + j];  // A(m=lm, k)

  #pragma unroll
  for (int nt = 0; nt < 9; ++nt) {
    v8f acc = {};
    #pragma unroll
    for (int kk3 = 0; kk3 < 3; ++kk3) {
      v2f b;
      #pragma unroll
      for (int j = 0; j < 2; ++j)
        b[j] = W2[(kk3*4 + 2*lhi + j)*144 + nt*16 + lm];   // always in-bounds
      acc = __builtin_amdgcn_wmma_f32_16x16x4_f32(false, aK[kk3], false, b, (short)0, acc, false, false);
    }
    #pragma unroll
    for (int g = 0; g < 8; ++g)
      ldsR[(g + 8*lhi)*144 + nt*16 + lm] = acc[g];
  }
  __syncthreads();
}

// ===========================================================================
// conv1 (path 0 x l -> l): scalar features in, C(0,l,l)=I/sqrt(2l+1)
// ===========================================================================
template<int L>
__global__ void __launch_bounds__(32) k_conv1(
    const float* __restrict__ edge_vec, const int* __restrict__ esrc,
    const int* __restrict__ edst,
    const float* __restrict__ W1, const float* __restrict__ bias,
    const float* __restrict__ W2,
    const float* __restrict__ feat, float* __restrict__ Aout)
{
  constexpr int NM = 2*L + 1;
  int lane = threadIdx.x;
  int lm = lane & 15, lhi = lane >> 4;
  __shared__ float ldsW1[20*16];
  __shared__ float ldsH[256];
  __shared__ float ldsR[16*144];

  int e = blockIdx.x*16 + lm;
  float ex = edge_vec[e*3+0], ey = edge_vec[e*3+1], ez = edge_vec[e*3+2];
  float r = sqrtf(ex*ex + ey*ey + ez*ez);

  radial_to_lds(lane, r, 20, 10.0f, W1, bias, W2, ldsW1, ldsH, ldsR);

  int src = esrc[e], dst = edst[e];
  float s[12];
  #pragma unroll
  for (int v = 0; v < 12; ++v) s[v] = feat[src*12 + v];

  float invr = 1.0f/(r + 1e-9f);
  float ux = ex*invr, uy = ey*invr, uz = ez*invr;
  float Y[NM];
  if constexpr (L == 0) { Y[0] = 0.28209479177387814f; }
  else if constexpr (L == 1) {
    Y[0] = 0.4886025119029199f*ux; Y[1] = 0.4886025119029199f*uy; Y[2] = 0.4886025119029199f*uz;
  } else {
    Y[0] = 0.5462742152960396f*(ux*ux - uy*uy);
    Y[1] = 1.0925484305920792f*ux*uy;
    Y[2] = 1.0925484305920792f*uy*uz;
    Y[3] = 0.31539156525252005f*(3.0f*uz*uz - 1.0f);
    Y[4] = 1.0925484305920792f*ux*uz;
  }
  const float scale = INV_DEG / sqrtf((float)(2*L + 1));

  #pragma unroll
  for (int uu = 0; uu < 6; ++uu) {
    int u = lhi*6 + uu;
    float t = 0.0f;
    #pragma unroll
    for (int v = 0; v < 12; ++v) t += ldsR[lm*144 + u*12 + v]*s[v];
    t *= scale;
    #pragma unroll
    for (int mo = 0; mo < NM; ++mo)
      atomicAdd(&Aout[(dst*12 + u)*NM + mo], t*Y[mo]);
  }
}

// ===========================================================================
// node middle stage: norm -> lin2 -> gated nonlin -> lin3  (per node, per l)
// ===========================================================================
template<int L>
__global__ void k_node_mid(const float* __restrict__ A,
                           const float* __restrict__ L2w,
                           const float* __restrict__ L3w,
                           float* __restrict__ Mo)
{
  constexpr int NM = 2*L + 1;
  int n = blockIdx.x*blockDim.x + threadIdx.x;
  if (n >= NNODES) return;
  float v[12*NM], w[12*NM];
  float ms = 0.0f;
  #pragma unroll
  for (int t = 0; t < 12*NM; ++t) { v[t] = A[(size_t)n*12*NM + t]; ms += v[t]*v[t]; }
  float inv = rsqrtf(ms * (1.0f/(12.0f*NM)) + 1e-8f);
  #pragma unroll
  for (int t = 0; t < 12*NM; ++t) v[t] *= inv;

  #pragma unroll
  for (int vc = 0; vc < 12; ++vc)
    #pragma unroll
    for (int m = 0; m < NM; ++m) {
      float acc = 0.0f;
      #pragma unroll
      for (int u = 0; u < 12; ++u) acc += v[u*NM + m]*L2w[u*12 + vc];
      w[vc*NM + m] = acc;
    }

  if constexpr (L == 0) {
    #pragma unroll
    for (int vc = 0; vc < 12; ++vc) w[vc] = sspf(w[vc]);
  } else {
    #pragma unroll
    for (int vc = 0; vc < 12; ++vc) {
      float nn = 1e-8f;
      #pragma unroll
      for (int m = 0; m < NM; ++m) nn += w[vc*NM+m]*w[vc*NM+m];
      nn = sqrtf(nn);
      float f = sspf(nn)/nn;
      #pragma unroll
      for (int m = 0; m < NM; ++m) w[vc*NM+m] *= f;
    }
  }

  #pragma unroll
  for (int vc = 0; vc < 12; ++vc)
    #pragma unroll
    for (int m = 0; m < NM; ++m) {
      float acc = 0.0f;
      #pragma unroll
      for (int u = 0; u < 12; ++u) acc += w[u*NM + m]*L3w[u*12 + vc];
      Mo[(size_t)n*12*NM + vc*NM + m] = acc;
    }
}

// ===========================================================================
// conv2: one kernel instantiation per (li,lf,lo) path
// ===========================================================================
template<int LI, int LF, int LO, int SLOT>
__global__ void __launch_bounds__(32) k_conv2(
    const float* __restrict__ edge_vec, const int* __restrict__ esrc,
    const int* __restrict__ edst,
    const float* __restrict__ W1, const float* __restrict__ bias,
    const float* __restrict__ W2,
    const float* __restrict__ Mi0, const float* __restrict__ Mi1,
    const float* __restrict__ Mi2,
    const float* __restrict__ Cg,
    float* __restrict__ B0o, float* __restrict__ B1o, float* __restrict__ B2o)
{
  constexpr int NI = 2*LI+1, NF = 2*LF+1, NO = 2*LO+1;
  constexpr int NODE_STRIDE = (LO==0) ? 36 : ((LO==1) ? 216 : 360);
  int lane = threadIdx.x;
  int lm = lane & 15, lhi = lane >> 4;
  __shared__ float ldsW1[40*16];
  __shared__ float ldsH[256];
  __shared__ float ldsR[16*144];
  __shared__ float ldsC[NI*NF*NO];

  for (int t = lane; t < NI*NF*NO; t += 32) ldsC[t] = Cg[t];

  int e = blockIdx.x*16 + lm;
  float ex = edge_vec[e*3+0], ey = edge_vec[e*3+1], ez = edge_vec[e*3+2];
  float r = sqrtf(ex*ex + ey*ey + ez*ez);

  radial_to_lds(lane, r, 40, 20.0f, W1, bias, W2, ldsW1, ldsH, ldsR);

  int src = esrc[e], dst = edst[e];
  const float* Min = (LI==0) ? Mi0 : ((LI==1) ? Mi1 : Mi2);
  float X[12*NI];
  #pragma unroll
  for (int t = 0; t < 12*NI; ++t) X[t] = Min[(size_t)src*12*NI + t];

  float invr = 1.0f/(r + 1e-9f);
  float ux = ex*invr, uy = ey*invr, uz = ez*invr;
  float Y[NF];
  if constexpr (LF == 0) { Y[0] = 0.28209479177387814f; }
  else if constexpr (LF == 1) {
    Y[0] = 0.4886025119029199f*ux; Y[1] = 0.4886025119029199f*uy; Y[2] = 0.4886025119029199f*uz;
  } else {
    Y[0] = 0.5462742152960396f*(ux*ux - uy*uy);
    Y[1] = 1.0925484305920792f*ux*uy;
    Y[2] = 1.0925484305920792f*uy*uz;
    Y[3] = 0.31539156525252005f*(3.0f*uz*uz - 1.0f);
    Y[4] = 1.0925484305920792f*ux*uz;
  }

  float P[NI*NO];
  #pragma unroll
  for (int mi = 0; mi < NI; ++mi)
    #pragma unroll
    for (int mo = 0; mo < NO; ++mo) {
      float acc = 0.0f;
      #pragma unroll
      for (int mf = 0; mf < NF; ++mf) acc += Y[mf]*ldsC[(mi*NF + mf)*NO + mo];
      P[mi*NO + mo] = acc;
    }

  float* Bout = (LO==0) ? B0o : ((LO==1) ? B1o : B2o);
  #pragma unroll
  for (int uu = 0; uu < 6; ++uu) {
    int u = lhi*6 + uu;
    float tv[NI];
    #pragma unroll
    for (int mi = 0; mi < NI; ++mi) {
      float a = 0.0f;
      #pragma unroll
      for (int v = 0; v < 12; ++v) a += ldsR[lm*144 + u*12 + v]*X[v*NI + mi];
      tv[mi] = a;
    }
    #pragma unroll
    for (int mo = 0; mo < NO; ++mo) {
      float m = 0.0f;
      #pragma unroll
      for (int mi = 0; mi < NI; ++mi) m += tv[mi]*P[mi*NO + mo];
      atomicAdd(&Bout[(size_t)dst*NODE_STRIDE + (SLOT*12 + u)*NO + mo], m*INV_DEG);
    }
  }
}

// ===========================================================================
// head: gather ca rows, norm/lin4/nonlin, 3-layer MLP, concat output [NCA,97]
// One wave per row.
// ===========================================================================
__global__ void __launch_bounds__(32) k_head(
    const int* __restrict__ ca_idx,
    const float* __restrict__ B0, const float* __restrict__ B1,
    const float* __restrict__ B2,
    const float* __restrict__ W0, const float* __restrict__ W1,
    const float* __restrict__ W2,
    const float* __restrict__ d1W, const float* __restrict__ d1b,
    const float* __restrict__ d2W, const float* __restrict__ d2b,
    const float* __restrict__ d3W, const float* __restrict__ d3b,
    float* __restrict__ out)
{
  int row = blockIdx.x;
  int lane = threadIdx.x;
  int n = ca_idx[row];
  __shared__ float s0[12];
  __shared__ float so1[36];
  __shared__ float so2[60];
  __shared__ float h1[250];
  __shared__ float h2[150];
  __shared__ float red[32];

  float a0 = B0[(size_t)n*36 + lane];               // lane < 32 < 36
  float a1 = (lane + 32 < 36) ? B0[(size_t)n*36 + lane + 32] : 0.0f;
  red[lane] = a0*a0 + a1*a1;
  __syncthreads();
  if (lane == 0) {
    float t = 0.0f;
    for (int k = 0; k < 32; ++k) t += red[k];
    red[0] = rsqrtf(t*(1.0f/36.0f) + 1e-8f);
  }
  __syncthreads();
  float inv = red[0];
  if (lane < 12) {
    float acc = 0.0f;
    for (int u = 0; u < 36; ++u) acc += B0[(size_t)n*36 + u]*W0[u*12 + lane];
    s0[lane] = sspf(acc*inv);
  }
  for (int idx = lane; idx < 36; idx += 32) {
    int v = idx/3, m = idx%3;
    float acc = 0.0f;
    for (int u = 0; u < 72; ++u) acc += B1[((size_t)n*72 + u)*3 + m]*W1[u*12 + v];
    so1[idx] = acc;
  }
  for (int idx = lane; idx < 60; idx += 32) {
    int v = idx/5, m = idx%5;
    float acc = 0.0f;
    for (int u = 0; u < 72; ++u) acc += B2[((size_t)n*72 + u)*5 + m]*W2[u*12 + v];
    so2[idx] = acc;
  }
  __syncthreads();
  if (lane < 12) {
    float nn = 1e-8f;
    for (int m = 0; m < 3; ++m) nn += so1[lane*3+m]*so1[lane*3+m];
    nn = sqrtf(nn);
    float f = sspf(nn)/nn;
    for (int m = 0; m < 3; ++m) so1[lane*3+m] *= f;
    nn = 1e-8f;
    for (int m = 0; m < 5; ++m) nn += so2[lane*5+m]*so2[lane*5+m];
    nn = sqrtf(nn);
    f = sspf(nn)/nn;
    for (int m = 0; m < 5; ++m) so2[lane*5+m] *= f;
  }
  __syncthreads();
  for (int j = lane; j < 250; j += 32) {
    float acc = d1b[j];
    for (int u = 0; u < 12; ++u) acc += s0[u]*d1W[u*250 + j];
    h1[j] = eluf(acc);
  }
  __syncthreads();
  for (int j = lane; j < 150; j += 32) {
    float acc = d2b[j];
    for (int k = 0; k < 250; ++k) acc += h1[k]*d2W[k*150 + j];
    h2[j] = eluf(acc);
  }
  __syncthreads();
  if (lane == 0) {
    float acc = d3b[0];
    for (int k = 0; k < 150; ++k) acc += h2[k]*d3W[k];
    out[(size_t)row*97] = acc;
  }
  for (int idx = lane; idx < 36; idx += 32) out[(size_t)row*97 + 1 + idx] = so1[idx];
  for (int idx = lane; idx < 60; idx += 32) out[(size_t)row*97 + 37 + idx] = so2[idx];
}

// ===========================================================================
extern "C" void kernel_launch(void* const* d_in, const int* in_sizes, int n_in,
                              void* d_out, int out_size, void* d_ws, size_t ws_size,
                              hipStream_t stream)
{
  (void)in_sizes; (void)n_in; (void)out_size; (void)ws_size;
  const float* x        = (const float*)d_in[0];
  const float* edge_vec = (const float*)d_in[1];
  const float* lin1_W   = (const float*)d_in[2];
  const float* c1_W1    = (const float*)d_in[3];
  const float* c1_b1    = (const float*)d_in[4];
  const float* c1_W2    = (const float*)d_in[5];
  const float* lin2_W   = (const float*)d_in[6];
  const float* lin3_W   = (const float*)d_in[7];
  const float* c2_W1    = (const float*)d_in[8];
  const float* c2_b1    = (const float*)d_in[9];
  const float* c2_W2    = (const float*)d_in[10];
  const float* lin4_W0  = (const float*)d_in[11];
  const float* lin4_W1  = (const float*)d_in[12];
  const float* lin4_W2  = (const float*)d_in[13];
  const float* d1_W     = (const float*)d_in[14];
  const float* d1_b     = (const float*)d_in[15];
  const float* d2_W     = (const float*)d_in[16];
  const float* d2_b     = (const float*)d_in[17];
  const float* d3_W     = (const float*)d_in[18];
  const float* d3_b     = (const float*)d_in[19];
  const int*   esrc     = (const int*)d_in[20];
  const int*   edst     = (const int*)d_in[21];
  const int*   ca_idx   = (const int*)d_in[22];
  float* out = (float*)d_out;
  float* wsf = (float*)d_ws;

  // zero atomic accumulators (harness does not re-poison between replays)
  hipMemsetAsync(wsf + OFF_A0, 0, (size_t)NNODES*108*sizeof(float), stream);
  hipMemsetAsync(wsf + OFF_B0, 0, (size_t)NNODES*612*sizeof(float), stream);

  k_cg_setup<<<1, 1, 0, stream>>>(wsf + OFF_CG);
  k_lin1<<<(NNODES*MULC + 255)/256, 256, 0, stream>>>(x, lin1_W, wsf + OFF_FEAT);

  const int TILES = NEDGES/16;
  k_conv1<0><<<TILES, 32, 0, stream>>>(edge_vec, esrc, edst, c1_W1,       c1_b1,    c1_W2,        wsf+OFF_FEAT, wsf+OFF_A0);
  k_conv1<1><<<TILES, 32, 0, stream>>>(edge_vec, esrc, edst, c1_W1+240,   c1_b1+12, c1_W2+1728,   wsf+OFF_FEAT, wsf+OFF_A1);
  k_conv1<2><<<TILES, 32, 0, stream>>>(edge_vec, esrc, edst, c1_W1+480,   c1_b1+24, c1_W2+3456,   wsf+OFF_FEAT, wsf+OFF_A2);

  k_node_mid<0><<<(NNODES+127)/128, 128, 0, stream>>>(wsf+OFF_A0, lin2_W,     lin3_W,     wsf+OFF_M0);
  k_node_mid<1><<<(NNODES+127)/128, 128, 0, stream>>>(wsf+OFF_A1, lin2_W+144, lin3_W+144, wsf+OFF_M1);
  k_node_mid<2><<<(NNODES+127)/128, 128, 0, stream>>>(wsf+OFF_A2, lin2_W+288, lin3_W+288, wsf+OFF_M2);

  #define LP(P, LI, LF, LO, SLOT, CGO)                                          \
    k_conv2<LI,LF,LO,SLOT><<<TILES, 32, 0, stream>>>(edge_vec, esrc, edst,      \
      c2_W1 + (P)*480, c2_b1 + (P)*12, c2_W2 + (P)*1728,                        \
      wsf+OFF_M0, wsf+OFF_M1, wsf+OFF_M2, wsf+OFF_CG+(CGO),                     \
      wsf+OFF_B0, wsf+OFF_B1, wsf+OFF_B2)

  LP(0, 0,0,0, 0, 0);    LP(1, 0,1,1, 0, 1);    LP(2, 0,2,2, 0, 10);
  LP(3, 1,0,1, 1, 35);   LP(4, 1,1,0, 1, 44);   LP(5, 1,1,1, 2, 53);
  LP(6, 1,1,2, 1, 80);   LP(7, 1,2,1, 3, 125);  LP(8, 1,2,2, 2, 170);
  LP(9, 2,0,2, 3, 245);  LP(10,2,1,1, 4, 270);  LP(11,2,1,2, 4, 315);
  LP(12,2,2,0, 2, 390);  LP(13,2,2,1, 5, 415);  LP(14,2,2,2, 5, 490);
  #undef LP

  k_head<<<NCA, 32, 0, stream>>>(ca_idx, wsf+OFF_B0, wsf+OFF_B1, wsf+OFF_B2,
      lin4_W0, lin4_W1, lin4_W2, d1_W, d1_b, d2_W, d2_b, d3_W, d3_b, out);
}